// ProteinFeatures_22033182228663
// MI455X (gfx1250) — compile-verified
//
#include <hip/hip_runtime.h>
#include <hip/hip_bf16.h>

// ---------------- problem constants ----------------
#define BB   2
#define NN   2048
#define KK   30
#define NRBF 16
#define CV   213           // V channels: 12*16 rbf + 12 angles + 9 direct
#define CE   480           // E channels: 29*16 rbf + 4 quat + 12 direct

// output layout (flat f32 elements)
#define V_OFF    ((size_t)0)
#define V_SZ     ((size_t)BB * NN * CV)                 // 872448
#define E_OFF    (V_OFF + V_SZ)
#define E_SZ     ((size_t)BB * NN * KK * CE)            // 58982400
#define EIDX_OFF (E_OFF + E_SZ)                         // 59854848

// workspace layout (flat f32 elements)
#define WS_ATOMS 0                                       // B*N*24 packed: N,Ca,C,O,Cb,v0,v1 (21 used)
#define WS_Q3    ((size_t)BB * NN * 24)                  // B*N*9 row-major frames
#define WS_XCAT  (WS_Q3 + (size_t)BB * NN * 9)           // B*4*N  SoA (comp-major, z-pad)
#define WS_N2    (WS_XCAT + (size_t)BB * 4 * NN)         // B*N    |Xca|^2
#define WS_DMAT  (WS_N2 + (size_t)BB * NN)               // B*N*N  distance matrix

typedef __attribute__((ext_vector_type(2))) float v2f;
typedef __attribute__((ext_vector_type(8))) float v8f;

struct f3 { float x, y, z; };
__device__ __forceinline__ f3 f3sub(f3 a, f3 b) { return {a.x-b.x, a.y-b.y, a.z-b.z}; }
__device__ __forceinline__ f3 f3add(f3 a, f3 b) { return {a.x+b.x, a.y+b.y, a.z+b.z}; }
__device__ __forceinline__ f3 f3scale(float s, f3 a) { return {s*a.x, s*a.y, s*a.z}; }
__device__ __forceinline__ float f3dot(f3 a, f3 b) { return a.x*b.x + a.y*b.y + a.z*b.z; }
__device__ __forceinline__ f3 f3cross(f3 a, f3 b) {
  return {a.y*b.z - a.z*b.y, a.z*b.x - a.x*b.z, a.x*b.y - a.y*b.x};
}
__device__ __forceinline__ f3 f3normalize(f3 a) {
  float n = fmaxf(sqrtf(f3dot(a, a)), 1e-8f);
  return {a.x/n, a.y/n, a.z/n};
}
__device__ __forceinline__ float sgnf(float x) { return (x > 0.f) ? 1.f : ((x < 0.f) ? -1.f : 0.f); }
__device__ __forceinline__ float clipf(float x, float lo, float hi) { return fminf(fmaxf(x, lo), hi); }

__device__ __forceinline__ f3 ldAtom(const float* Xb, int n, int a) {
  const float* p = Xb + ((size_t)n * 4 + a) * 3;
  return {p[0], p[1], p[2]};
}
// X3 chain element t in [0,3N): atom t%3 of residue t/3
__device__ __forceinline__ f3 ldChain(const float* Xb, int t) { return ldAtom(Xb, t / 3, t % 3); }

// RBF constants: mu_m = 20*m/15, sigma = 20/16 = 1.25
#define RBF_MU(m)  ((20.0f / 15.0f) * (float)(m))
#define RBF_ISIG   (1.0f / 1.25f)

// atom indices in packed table: 0=N 1=Ca 2=C 3=O 4=Cb 5=vat0 6=vat1
__constant__ int c_npf[12] = {1,1,1,0,0,3,1,4,4,2,6,5};
__constant__ int c_nps[12] = {0,2,3,2,3,2,4,0,3,4,5,6};
__constant__ int c_epf[29] = {1,1,2,1,0,4,1,4,0,4,3,4,2,4,1,3,2,2,0,2,3,0,0,3,3,5,6,6,5};
__constant__ int c_eps[29] = {1,2,1,0,1,1,4,0,4,3,4,2,4,4,3,1,2,0,2,3,2,0,3,0,3,5,6,5,6};
__constant__ int c_ord[4]  = {1,0,2,3};   // neighbor atom order for E_direct: Ca,N,C,O

// ======================= kernel 1: per-node features =======================
__global__ void k_node(const float* __restrict__ X, const float* __restrict__ vatm,
                       float* __restrict__ atoms, float* __restrict__ q3ws,
                       float* __restrict__ xcaT, float* __restrict__ n2,
                       float* __restrict__ V) {
  int gid = blockIdx.x * blockDim.x + threadIdx.x;
  if (gid >= BB * NN) return;
  int b = gid / NN, n = gid % NN;
  const float* Xb = X + (size_t)b * NN * 4 * 3;

  f3 aN = ldAtom(Xb, n, 0), aCa = ldAtom(Xb, n, 1);
  f3 aC = ldAtom(Xb, n, 2), aO  = ldAtom(Xb, n, 3);

  // virtual Cb and virtual atoms
  f3 bv = f3sub(aCa, aN), cv = f3sub(aC, aCa), av = f3cross(bv, cv);
  f3 aCb = f3add(f3add(f3scale(-0.58273431f, av), f3scale(0.56802827f, bv)),
                 f3add(f3scale(-0.54067466f, cv), aCa));
  f3 vat[2];
  #pragma unroll
  for (int i = 0; i < 2; i++) {
    f3 va = {vatm[i*3 + 0], vatm[i*3 + 1], vatm[i*3 + 2]};
    float nr = sqrtf(f3dot(va, va));
    va = {va.x / nr, va.y / nr, va.z / nr};
    vat[i] = f3add(f3add(f3scale(va.x, av), f3scale(va.y, bv)),
                   f3add(f3scale(va.z, cv), aCa));
  }

  f3 at[7] = {aN, aCa, aC, aO, aCb, vat[0], vat[1]};
  float* ap = atoms + (size_t)gid * 24;
  #pragma unroll
  for (int i = 0; i < 7; i++) { ap[i*3+0] = at[i].x; ap[i*3+1] = at[i].y; ap[i*3+2] = at[i].z; }

  // SoA-transposed Ca (z-padded 4th comp) for WMMA B-frags; squared norms
  xcaT[((size_t)b*4 + 0)*NN + n] = aCa.x;
  xcaT[((size_t)b*4 + 1)*NN + n] = aCa.y;
  xcaT[((size_t)b*4 + 2)*NN + n] = aCa.z;
  xcaT[((size_t)b*4 + 3)*NN + n] = 0.f;
  n2[gid] = f3dot(aCa, aCa);

  // local frame Q3 = [b1; n0; b1 x n0], zero for last node (padding in reference)
  f3 r0 = {0,0,0}, r1 = {0,0,0}, r2 = {0,0,0};
  if (n < NN - 1) {
    f3 u0q = f3normalize(f3sub(aCa, aN));
    f3 u1q = f3normalize(f3sub(aC, aCa));
    r1 = f3normalize(f3cross(u0q, u1q));
    r0 = f3normalize(f3sub(u0q, u1q));
    r2 = f3cross(r0, r1);
  }
  float* qp = q3ws + (size_t)gid * 9;
  qp[0]=r0.x; qp[1]=r0.y; qp[2]=r0.z;
  qp[3]=r1.x; qp[4]=r1.y; qp[5]=r1.z;
  qp[6]=r2.x; qp[7]=r2.y; qp[8]=r2.z;

  float* Vp = V + (size_t)gid * CV;

  // ---- node RBFs (12 pairs x 16) ----
  #pragma unroll
  for (int p = 0; p < 12; p++) {
    f3 d3 = f3sub(at[c_npf[p]], at[c_nps[p]]);
    float d = sqrtf(f3dot(d3, d3) + 1e-6f);
    #pragma unroll
    for (int m = 0; m < NRBF; m++) {
      float t = (d - RBF_MU(m)) * RBF_ISIG;
      Vp[p * NRBF + m] = expf(-t * t);
    }
  }

  // ---- dihedrals / bond angles (padded like the reference) ----
  const int tN = 3 * NN;
  #pragma unroll
  for (int s = 0; s < 3; s++) {
    int i = 3 * n + s - 1;
    float Dang = 0.f, Aang = 0.f;
    if (i >= 0 && i <= tN - 4) {
      f3 u0 = f3normalize(f3sub(ldChain(Xb, i+1), ldChain(Xb, i)));
      f3 u1 = f3normalize(f3sub(ldChain(Xb, i+2), ldChain(Xb, i+1)));
      f3 u2 = f3normalize(f3sub(ldChain(Xb, i+3), ldChain(Xb, i+2)));
      f3 n0 = f3normalize(f3cross(u0, u1));
      f3 n1 = f3normalize(f3cross(u1, u2));
      float cd = clipf(f3dot(n0, n1), -1.f + 1e-7f, 1.f - 1e-7f);
      f3 vv = f3normalize(f3cross(n0, n1));
      Dang = sgnf(-f3dot(vv, u1)) * acosf(cd);
      Aang = acosf(clipf(f3dot(u0, u1), -1.f + 1e-7f, 1.f - 1e-7f));
    }
    Vp[192 + s] = cosf(Dang);
    Vp[195 + s] = sinf(Dang);
    Vp[198 + s] = cosf(Aang);
    Vp[201 + s] = sinf(Aang);
  }

  // ---- V_direct: normalize(Q3 * (atom_{0,2,3} - N)) ----
  const int innerIdx[3] = {0, 2, 3};
  #pragma unroll
  for (int t = 0; t < 3; t++) {
    f3 d3 = f3sub(at[innerIdx[t]], aN);
    f3 u = {f3dot(r0, d3), f3dot(r1, d3), f3dot(r2, d3)};
    u = f3normalize(u);
    Vp[204 + t*3 + 0] = u.x; Vp[204 + t*3 + 1] = u.y; Vp[204 + t*3 + 2] = u.z;
  }
}

// ======== kernel 2a: pairwise distance matrix via V_WMMA_F32_16X16X4_F32 ========
// one wave per 16-row strip; 128 col-tiles of 16; D stays L2-resident (33.5 MB < 192 MB)
__global__ void k_dist(const float* __restrict__ xcaT, const float* __restrict__ n2,
                       const float* __restrict__ mask, float* __restrict__ Dmat) {
  int wave = (blockIdx.x * blockDim.x + threadIdx.x) >> 5;   // 0..255, exact
  int lane = threadIdx.x & 31;
  int b = wave >> 7;
  int strip = wave & 127;
  int half = lane >> 4, l16 = lane & 15;
  int kb = half * 2;                       // K components held by this lane half
  const float* xT = xcaT + (size_t)b * 4 * NN;

  // A-frag (16x4 f32): lane M = l16; VGPR0=K kb, VGPR1=K kb+1
  int row = strip * 16 + l16;
  v2f a;
  a.x = xT[(size_t)kb * NN + row];
  a.y = xT[(size_t)(kb + 1) * NN + row];

  float n2row[8], mrow[8];
  #pragma unroll
  for (int v = 0; v < 8; v++) {
    int rM = strip * 16 + v + 8 * half;    // C/D layout: VGPR v -> M = v + 8*half
    n2row[v] = n2[(size_t)b * NN + rM];
    mrow[v]  = mask[(size_t)b * NN + rM];
  }

  for (int tile = 0; tile < NN / 16; tile++) {
    int col = tile * 16 + l16;
    v2f bf;                                // B-frag (4x16 f32): lane N = l16, same K split
    bf.x = xT[(size_t)kb * NN + col];
    bf.y = xT[(size_t)(kb + 1) * NN + col];
    if (tile + 1 < NN / 16)
      __builtin_prefetch(&xT[(size_t)kb * NN + col + 16], 0, 3);  // global_prefetch_b8

    v8f c = {};
    c = __builtin_amdgcn_wmma_f32_16x16x4_f32(
        /*neg_a=*/false, a, /*neg_b=*/false, bf,
        /*c_mod=*/(short)0, c, /*reuse_a=*/false, /*reuse_b=*/false);

    float n2c = n2[(size_t)b * NN + col];
    float mc  = mask[(size_t)b * NN + col];
    #pragma unroll
    for (int v = 0; v < 8; v++) {
      float d2 = n2row[v] + n2c - 2.f * c[v];
      float m2 = mrow[v] * mc;
      float Dv = (1.f - m2) * 10000.f + m2 * sqrtf(fmaxf(d2, 0.f) + 1e-6f);
      Dmat[((size_t)b * NN + strip * 16 + v + 8 * half) * NN + col] = Dv;
    }
  }
}

// ================= kernel 2b: top-k (k=30) smallest per row =================
__global__ void k_topk(const float* __restrict__ Dmat, int* __restrict__ Eidx) {
  const int tid = threadIdx.x;
  const int row = blockIdx.x;                        // b*N+n
  const float* Drow = Dmat + (size_t)row * NN;
  __shared__ float sval[256];
  __shared__ int   sidx[256];

  float vals[NN / 256];
  #pragma unroll
  for (int q = 0; q < NN / 256; q++) vals[q] = Drow[tid + q * 256];

  for (int kk = 0; kk < KK; kk++) {
    float mv = 3.402823466e38f; int mi = NN;
    #pragma unroll
    for (int q = 0; q < NN / 256; q++) {
      int col = tid + q * 256;
      if (vals[q] < mv || (vals[q] == mv && col < mi)) { mv = vals[q]; mi = col; }
    }
    sval[tid] = mv; sidx[tid] = mi;
    __syncthreads();
    for (int s = 128; s > 0; s >>= 1) {
      if (tid < s) {
        float ov = sval[tid + s]; int oi = sidx[tid + s];
        if (ov < sval[tid] || (ov == sval[tid] && oi < sidx[tid])) { sval[tid] = ov; sidx[tid] = oi; }
      }
      __syncthreads();
    }
    int widx = sidx[0];
    if (tid == 0) Eidx[(size_t)row * KK + kk] = widx;
    if ((widx & 255) == tid) vals[widx >> 8] = 3.402823466e38f;   // invalidate
    __syncthreads();
  }
}

// ================= kernel 3: per-edge features (one wave32 per edge) =================
__global__ void k_edge(const float* __restrict__ atoms, const float* __restrict__ q3ws,
                       const int* __restrict__ Eidx, float* __restrict__ E) {
  size_t wid = (size_t)blockIdx.x * (blockDim.x >> 5) + (threadIdx.x >> 5);
  int lane = threadIdx.x & 31;
  int b = (int)(wid / ((size_t)NN * KK));
  int rem = (int)(wid % ((size_t)NN * KK));
  int n = rem / KK;
  int j = Eidx[wid];

  const float* ai = atoms + ((size_t)b * NN + n) * 24;
  const float* aj = atoms + ((size_t)b * NN + j) * 24;
  const float* Qi = q3ws + ((size_t)b * NN + n) * 9;
  const float* Qj = q3ws + ((size_t)b * NN + j) * 9;
  float* Ee = E + wid * CE;

  // lanes 0..28: one pair distance each
  float dist = 0.f;
  if (lane < 29) {
    const float* pa = ai + c_epf[lane] * 3;
    const float* pb = aj + c_eps[lane] * 3;
    float dx = pa[0]-pb[0], dy = pa[1]-pb[1], dz = pa[2]-pb[2];
    dist = sqrtf(dx*dx + dy*dy + dz*dz + 1e-6f);
  }

  // 464 RBF channels; pair distance broadcast by lane shuffle
  for (int c = lane; c < 29 * NRBF; c += 32) {
    int p = c >> 4, m = c & 15;
    float d = __shfl(dist, p, 32);
    float t = (d - RBF_MU(m)) * RBF_ISIG;
    Ee[c] = expf(-t * t);
  }

  // lanes 0..3: quaternion of R = Qi^T * Qj
  if (lane < 4) {
    float R[3][3];
    #pragma unroll
    for (int ii = 0; ii < 3; ii++)
      #pragma unroll
      for (int ll = 0; ll < 3; ll++) {
        float s = 0.f;
        #pragma unroll
        for (int jj = 0; jj < 3; jj++) s += Qi[jj*3 + ii] * Qj[jj*3 + ll];
        R[ii][ll] = s;
      }
    float Rxx = R[0][0], Ryy = R[1][1], Rzz = R[2][2];
    float q0 = sgnf(R[2][1] - R[1][2]) * 0.5f * sqrtf(fabsf(1.f + Rxx - Ryy - Rzz));
    float q1 = sgnf(R[0][2] - R[2][0]) * 0.5f * sqrtf(fabsf(1.f - Rxx + Ryy - Rzz));
    float q2 = sgnf(R[1][0] - R[0][1]) * 0.5f * sqrtf(fabsf(1.f - Rxx - Ryy + Rzz));
    float w  = 0.5f * sqrtf(fmaxf(1.f + Rxx + Ryy + Rzz, 0.f));
    float qn = fmaxf(sqrtf(q0*q0 + q1*q1 + q2*q2 + w*w), 1e-8f);
    float q4 = (lane == 0) ? q0 : (lane == 1) ? q1 : (lane == 2) ? q2 : w;
    Ee[464 + lane] = q4 / qn;
  }

  // lanes 4..15: E_direct = normalize(Qi * (neighbor atom[Ca,N,C,O] - N_i))
  if (lane >= 4 && lane < 16) {
    int c2 = lane - 4, a = c2 / 3, comp = c2 % 3;
    const float* pj = aj + c_ord[a] * 3;
    float dx = pj[0] - ai[0], dy = pj[1] - ai[1], dz = pj[2] - ai[2];
    float vx = Qi[0]*dx + Qi[1]*dy + Qi[2]*dz;
    float vy = Qi[3]*dx + Qi[4]*dy + Qi[5]*dz;
    float vz = Qi[6]*dx + Qi[7]*dy + Qi[8]*dz;
    float nrm = fmaxf(sqrtf(vx*vx + vy*vy + vz*vz), 1e-8f);
    float val = (comp == 0 ? vx : comp == 1 ? vy : vz) / nrm;
    Ee[468 + c2] = val;
  }
}

// =============================== launcher ===============================
extern "C" void kernel_launch(void* const* d_in, const int* in_sizes, int n_in,
                              void* d_out, int out_size, void* d_ws, size_t ws_size,
                              hipStream_t stream) {
  const float* X    = (const float*)d_in[0];   // (B,N,4,3)
  const float* mask = (const float*)d_in[1];   // (B,N)
  const float* vatm = (const float*)d_in[2];   // (VNUM,3)

  float* ws    = (float*)d_ws;
  float* atoms = ws + WS_ATOMS;
  float* q3ws  = ws + WS_Q3;
  float* xcaT  = ws + WS_XCAT;
  float* n2    = ws + WS_N2;
  float* Dmat  = ws + WS_DMAT;

  float* out   = (float*)d_out;
  float* Vout  = out + V_OFF;
  float* Eout  = out + E_OFF;
  int*   Eidx  = (int*)(out + EIDX_OFF);

  // 1) per-node features + staging (4096 threads)
  k_node<<<(BB * NN) / 256, 256, 0, stream>>>(X, vatm, atoms, q3ws, xcaT, n2, Vout);

  // 2a) WMMA distance matrix: 256 waves (one per 16-row strip)
  k_dist<<<32, 256, 0, stream>>>(xcaT, n2, mask, Dmat);

  // 2b) top-30 per row
  k_topk<<<BB * NN, 256, 0, stream>>>(Dmat, Eidx);

  // 3) per-edge features: one wave32 per edge, 8 waves/block
  k_edge<<<(BB * NN * KK) / 8, 256, 0, stream>>>(atoms, q3ws, Eidx, Eout);
}